// GATedge_90752658965188
// MI455X (gfx1250) — compile-verified
//
#include <hip/hip_runtime.h>

typedef __attribute__((ext_vector_type(2))) float v2f;
typedef __attribute__((ext_vector_type(8))) float v8f;
typedef int v4i __attribute__((vector_size(16)));
typedef __attribute__((address_space(1))) v4i* gas_v4i;   /* global int4*  */
typedef __attribute__((address_space(3))) v4i* las_v4i;   /* LDS int4*     */

#define NB      8
#define N_OP    1000
#define N_MA    64
#define IN_SRC  128
#define IN_DST  64
#define DF      128
#define CHUNK   128
#define NCHUNK  8      /* ceil(1000/128) */
#define TSTR    132    /* padded stride for t[64][128] in LDS */
#define NEG_SLOPE 0.2f
#define HSTRIDE 16384  /* floats between h_src double buffers */
#define ASTRIDE 8448   /* floats between alpha double buffers */

/* ---- async global->LDS support (gfx1250 GLOBAL_LOAD_ASYNC_TO_LDS_*) ---- */
#if defined(__has_builtin)
#  if __has_builtin(__builtin_amdgcn_global_load_async_to_lds_b128)
#    define HAVE_ASYNC 1
#  else
#    define HAVE_ASYNC 0
#  endif
#else
#  define HAVE_ASYNC 0
#endif

__device__ __forceinline__ void async_cp16(const float* g, float* l) {
#if HAVE_ASYNC
    __builtin_amdgcn_global_load_async_to_lds_b128(
        (gas_v4i)(g), (las_v4i)(l), 0, 0);
#else
    *reinterpret_cast<float4*>(l) = *reinterpret_cast<const float4*>(g);
#endif
}

__device__ __forceinline__ void async_wait_all() {
#if HAVE_ASYNC
#  if __has_builtin(__builtin_amdgcn_s_wait_asynccnt)
    __builtin_amdgcn_s_wait_asynccnt(0);
#  else
    asm volatile("s_wait_asynccnt 0x0" ::: "memory");
#  endif
#endif
}

/* ---- LDS layout (in floats) ---- */
#define OFF_H0    0                    /* 16384: h_src chunk buffer 0, later W_src */
#define OFF_H1    16384                /* 16384: h_src chunk buffer 1 */
#define OFF_A0    32768                /* 8448 : alpha buffer 0, later t [64][TSTR] */
#define OFF_A1    (OFF_A0 + ASTRIDE)   /* 8448 : alpha buffer 1 */
#define OFF_FD    (OFF_A1 + ASTRIDE)   /* 8192 : feat_dst [64][128] */
#define OFF_EL    (OFF_FD + 8192)      /* 1024 : el[1000] */
#define OFF_VL    (OFF_EL + 1024)      /* 128  : W_src @ attn_l */
#define OFF_ER    (OFF_VL + 128)       /* 64 */
#define OFF_MX    (OFF_ER + 64)        /* 64 : softmax max */
#define OFF_IZ    (OFF_MX + 64)        /* 64 : 1/Z */
#define OFF_SE    (OFF_IZ + 64)        /* 64 : S_e accumulators */
#define OFF_AKK   (OFF_SE + 64)        /* 64 : alpha_kk */
#define OFF_PMX   (OFF_AKK + 64)       /* 256: per-group partial max */
#define OFF_PS    (OFF_PMX + 256)      /* 256: per-group partial sum */
#define OFF_SWE   (OFF_PS + 256)       /* 4  : s_we scalar */
#define SMEM_FLOATS (OFF_SWE + 4)      /* ~59844 floats = 239376 B < 320 KB */

/* stage one 128x128 h_src chunk into an LDS buffer (async, no VGPR staging) */
__device__ __forceinline__ void load_hsrc_chunk(int b, int o0, int tid,
                                                const float* __restrict__ h_src,
                                                float* hbuf) {
    for (int idx = tid; idx < (CHUNK * IN_SRC) / 4; idx += 256) {
        const int r = idx >> 5, c4 = (idx & 31) << 2, o = o0 + r;
        if (o < N_OP)
            async_cp16(h_src + ((size_t)(b * N_OP + o) * IN_SRC + c4),
                       hbuf + r * IN_SRC + c4);
        /* o >= N_OP: leave stale (finite) data; alpha==0 nullifies those rows */
    }
}

/* recompute one 128x64 alpha chunk into LDS, accumulate S_e */
__device__ __forceinline__ void compute_alpha_chunk(
    int b, int o0, int tid,
    const float* __restrict__ edge, const int* __restrict__ adj,
    const float* sh_el, const float* sh_er, const float* sh_mx,
    const float* sh_iz, const float* sh_swe, float* abuf, float* sh_se) {
    const int m = tid & 63, grp = tid >> 6;
    const float swe = sh_swe[0], erm = sh_er[m], mxm = sh_mx[m], izm = sh_iz[m];
    float sel = 0.f;
    for (int r = grp; r < CHUNK; r += 4) {
        const int o = o0 + r;
        float al = 0.f;
        if (o < N_OP && adj[((size_t)b * N_OP + o) * N_MA + m] == 1) {
            const float ef = edge[((size_t)b * N_OP + o) * N_MA + m];
            float a = sh_el[o] + ef * swe + erm;
            float e = a > 0.f ? a : NEG_SLOPE * a;
            al = __expf(e - mxm) * izm;
            sel += ef * al;
        }
        abuf[r * 64 + m] = al;
    }
    atomicAdd(&sh_se[m], sel);
}

__global__ __launch_bounds__(256)
void gat_fused(const float* __restrict__ h_src, const float* __restrict__ h_dst,
               const float* __restrict__ edge,  const int*   __restrict__ adj,
               const float* __restrict__ W_src, const float* __restrict__ W_dst,
               const float* __restrict__ W_edge,const float* __restrict__ attn_l,
               const float* __restrict__ attn_r,float* __restrict__ out)
{
    extern __shared__ float sm[];
    const int b    = blockIdx.x;
    const int tid  = threadIdx.x;
    const int lane = tid & 31;
    const int wave = tid >> 5;

    float* sh_h0  = sm + OFF_H0;
    float* sh_a0  = sm + OFF_A0;
    float* sh_fd  = sm + OFF_FD;
    float* sh_el  = sm + OFF_EL;
    float* sh_vl  = sm + OFF_VL;
    float* sh_er  = sm + OFF_ER;
    float* sh_mx  = sm + OFF_MX;
    float* sh_iz  = sm + OFF_IZ;
    float* sh_se  = sm + OFF_SE;
    float* sh_akk = sm + OFF_AKK;
    float* sh_pmx = sm + OFF_PMX;
    float* sh_ps  = sm + OFF_PS;
    float* sh_swe = sm + OFF_SWE;

    /* ---- phase 0: v_l = W_src@attn_l ; s_we ; feat_dst = h_dst@W_dst ---- */
    if (tid < IN_SRC) {
        float a = 0.f;
        for (int d = 0; d < DF; ++d) a += W_src[tid * DF + d] * attn_l[d];
        sh_vl[tid] = a;
    }
    if (tid == 0) {
        float a = 0.f;
        for (int d = 0; d < DF; ++d) a += W_edge[d] * attn_l[d];
        sh_swe[0] = a;
    }
    for (int idx = tid; idx < N_MA * DF; idx += 256) {
        const int m = idx >> 7, d = idx & 127;
        const float* hr = h_dst + (size_t)(b * N_MA + m) * IN_DST;
        float a = 0.f;
        for (int k = 0; k < IN_DST; ++k) a += hr[k] * W_dst[k * DF + d];
        sh_fd[idx] = a;
    }
    __syncthreads();

    /* ---- phase 1: er ; el[o] = h_src[o] . v_l ---- */
    if (tid < N_MA) {
        float a = 0.f;
        for (int d = 0; d < DF; ++d) a += sh_fd[tid * DF + d] * attn_r[d];
        sh_er[tid] = a;
    }
    for (int o = tid; o < N_OP; o += 256) {
        const float4* r4 = reinterpret_cast<const float4*>(h_src + (size_t)(b * N_OP + o) * IN_SRC);
        float a = 0.f;
        for (int k = 0; k < IN_SRC / 4; ++k) {
            float4 v = r4[k];
            a += v.x * sh_vl[4*k] + v.y * sh_vl[4*k+1] + v.z * sh_vl[4*k+2] + v.w * sh_vl[4*k+3];
        }
        sh_el[o] = a;
    }
    __syncthreads();

    /* ---- phase 2: online softmax stats over o-axis (per m) ---- */
    {
        const int m = tid & 63, grp = tid >> 6;
        const float swe = sh_swe[0], erm = sh_er[m];
        float mx = -__builtin_inff(), s = 0.f;
        for (int o = grp; o < N_OP; o += 4) {
            if (adj[((size_t)b * N_OP + o) * N_MA + m] == 1) {
                float a = sh_el[o] + edge[((size_t)b * N_OP + o) * N_MA + m] * swe + erm;
                float e = a > 0.f ? a : NEG_SLOPE * a;
                float nmx = fmaxf(mx, e);
                s = s * __expf(mx - nmx) + __expf(e - nmx);
                mx = nmx;
            }
        }
        sh_pmx[grp * 64 + m] = mx;
        sh_ps [grp * 64 + m] = s;
    }
    __syncthreads();
    if (tid < N_MA) {
        const int m = tid;
        const float er2 = sh_er[m] * 2.f;
        const float ekk = er2 > 0.f ? er2 : NEG_SLOPE * er2;
        float mx = ekk, s = 1.f;                     /* kk row always unmasked */
        for (int g = 0; g < 4; ++g) {
            float pm = sh_pmx[g * 64 + m], ps = sh_ps[g * 64 + m];
            float nmx = fmaxf(mx, pm);
            s  = s * __expf(mx - nmx) + ps * __expf(pm - nmx);
            mx = nmx;
        }
        const float iz = 1.f / s;
        sh_mx[m] = mx; sh_iz[m] = iz;
        sh_akk[m] = __expf(ekk - mx) * iz;
        sh_se[m]  = 0.f;
    }
    __syncthreads();

    /* ---- phase 3: t = alpha^T @ h_src, double-buffered async chunks ---- */
    const int mt = wave & 3;          /* M-tile of this wave (m dim, 4 tiles)      */
    const int nbase = (wave >> 2)*4;  /* first of 4 N-tiles (k dim, 8 tiles total) */
    const int lh = lane & 15;
    const int hi = lane >> 4;         /* K half: lanes 16-31 hold K=2,3            */

    /* prologue: chunk 0 in flight + alpha 0 */
    load_hsrc_chunk(b, 0, tid, h_src, sh_h0);
    compute_alpha_chunk(b, 0, tid, edge, adj, sh_el, sh_er, sh_mx, sh_iz,
                        sh_swe, sh_a0, sh_se);
    async_wait_all();
    __syncthreads();

    v8f acc[4] = {};
    for (int c = 0; c < NCHUNK; ++c) {
        /* kick off next chunk's DMA before touching the matrix pipe.
           Buffer selection is pure offset arithmetic on the LDS base so the
           address space stays inferable (ds_load, not flat_load). */
        if (c + 1 < NCHUNK)
            load_hsrc_chunk(b, (c + 1) * CHUNK, tid, h_src,
                            sh_h0 + ((c + 1) & 1) * HSTRIDE);

        const float* hbuf = sh_h0 + (c & 1) * HSTRIDE;
        const float* abuf = sh_a0 + (c & 1) * ASTRIDE;
        for (int ks = 0; ks < CHUNK / 4; ++ks) {
            const int ob = ks * 4 + hi * 2;
            v2f A;
            A.x = abuf[ ob      * 64 + mt * 16 + lh];   /* alpha^T[m, o] */
            A.y = abuf[(ob + 1) * 64 + mt * 16 + lh];
            #pragma unroll
            for (int j = 0; j < 4; ++j) {
                const int n = (nbase + j) * 16 + lh;
                v2f Bv;
                Bv.x = hbuf[ ob      * IN_SRC + n];     /* h_src[o, k] */
                Bv.y = hbuf[(ob + 1) * IN_SRC + n];
                acc[j] = __builtin_amdgcn_wmma_f32_16x16x4_f32(
                    false, A, false, Bv, (short)0, acc[j], false, false);
            }
        }

        /* next chunk's alpha on the VALU while WMMAs drain */
        if (c + 1 < NCHUNK)
            compute_alpha_chunk(b, (c + 1) * CHUNK, tid, edge, adj, sh_el,
                                sh_er, sh_mx, sh_iz, sh_swe,
                                sh_a0 + ((c + 1) & 1) * ASTRIDE, sh_se);

        async_wait_all();
        __syncthreads();
    }

    /* ---- phase 4: spill t -> LDS (padded), async-stage W_src ---- */
    float* sh_t = sh_a0;
    #pragma unroll
    for (int j = 0; j < 4; ++j) {
        #pragma unroll
        for (int r = 0; r < 8; ++r) {
            const int m = mt * 16 + hi * 8 + r;
            const int k = (nbase + j) * 16 + lh;
            sh_t[m * TSTR + k] = acc[j][r];
        }
    }
    for (int idx = tid; idx < (IN_SRC * DF) / 4; idx += 256)
        async_cp16(W_src + idx * 4, sh_h0 + idx * 4);
    async_wait_all();
    __syncthreads();

    /* ---- phase 5: out_part = t @ W_src (K = 128) ---- */
    v8f oacc[4] = {};
    for (int ks = 0; ks < IN_SRC / 4; ++ks) {
        const int kb = ks * 4 + hi * 2;
        v2f A;
        A.x = sh_t[(mt * 16 + lh) * TSTR + kb];
        A.y = sh_t[(mt * 16 + lh) * TSTR + kb + 1];
        #pragma unroll
        for (int j = 0; j < 4; ++j) {
            const int n = (nbase + j) * 16 + lh;
            v2f Bv;
            Bv.x = sh_h0[ kb      * DF + n];
            Bv.y = sh_h0[(kb + 1) * DF + n];
            oacc[j] = __builtin_amdgcn_wmma_f32_16x16x4_f32(
                false, A, false, Bv, (short)0, oacc[j], false, false);
        }
    }

    /* ---- epilogue: sigmoid(b_part + W_edge*S_e + feat_dst*alpha_kk) ---- */
    #pragma unroll
    for (int j = 0; j < 4; ++j) {
        const int d = (nbase + j) * 16 + lh;
        const float we = W_edge[d];
        #pragma unroll
        for (int r = 0; r < 8; ++r) {
            const int m = mt * 16 + hi * 8 + r;
            const float val = oacc[j][r] + we * sh_se[m] + sh_fd[m * DF + d] * sh_akk[m];
            out[((size_t)b * N_MA + m) * DF + d] = 1.f / (1.f + __expf(-val));
        }
    }
}

extern "C" void kernel_launch(void* const* d_in, const int* in_sizes, int n_in,
                              void* d_out, int out_size, void* d_ws, size_t ws_size,
                              hipStream_t stream) {
    const float* h_src  = (const float*)d_in[0];
    const float* h_dst  = (const float*)d_in[1];
    const float* edge   = (const float*)d_in[2];
    const int*   adj    = (const int*)  d_in[3];
    const float* W_src  = (const float*)d_in[4];
    const float* W_dst  = (const float*)d_in[5];
    const float* W_edge = (const float*)d_in[6];
    const float* attn_l = (const float*)d_in[7];
    const float* attn_r = (const float*)d_in[8];
    float* out = (float*)d_out;

    const size_t smem = (size_t)SMEM_FLOATS * sizeof(float);
    gat_fused<<<dim3(NB), dim3(256), smem, stream>>>(
        h_src, h_dst, edge, adj, W_src, W_dst, W_edge, attn_l, attn_r, out);
}